// CRF_69647189672048
// MI455X (gfx1250) — compile-verified
//
#include <hip/hip_runtime.h>

typedef float v2f __attribute__((ext_vector_type(2)));
typedef float v8f __attribute__((ext_vector_type(8)));

#define S_LEN 2048
#define B_SZ  256
#define T_SZ  48

// LDS strides: multiples of 4 (float4-aligned phase-A accesses); 52 keeps the
// A-operand reads (bank = (52*l + k) & 63) and C-tile writes conflict-free.
#define SPAD 52
#define PPAD 52

#define LOG2E 1.44269504088896340736f
#define LN2   0.69314718055994530942f

// Raw hardware transcendentals: inputs are guaranteed non-denormal here
// (exp2 args <= 0 -> clean underflow; log args >= exp(-0.1) ~= 0.905).
static __device__ __forceinline__ float fast_exp(float x) {
    return __builtin_amdgcn_exp2f(x * LOG2E);
}
static __device__ __forceinline__ float fast_log(float x) {
    return __builtin_amdgcn_logf(x) * LN2;
}

// ---------------------------------------------------------------------------
// Normalizer (forward algorithm) with exp-space WMMA GEMM.
// grid = 16 (batch tiles of 16 rows), block = 96 (3 waves; wave w = col tile w)
// Per step: phase A (combine maxes, build P = exp(S-m)) | barrier |
//           phase B (12x v_wmma_f32_16x16x4 in two chains, epilogue with
//                    register-carried old scores, in-register row-max via
//                    half-wave shuffles) | barrier.
// ---------------------------------------------------------------------------
__global__ __launch_bounds__(96) void crf_forward_norm_kernel(
    const float* __restrict__ emissions,   // (S,B,T)
    const int*   __restrict__ mask,        // (S,B) int32
    const float* __restrict__ start_tr,    // (T)
    const float* __restrict__ end_tr,      // (T)
    const float* __restrict__ trans,       // (T,T)
    float* __restrict__ den)               // (B)
{
    __shared__ __attribute__((aligned(16))) float lds_S[16 * SPAD]; // scores
    __shared__ __attribute__((aligned(16))) float lds_P[16 * PPAD]; // exp(S-m)
    __shared__ float lds_pm[16 * 4];   // per-row partial maxes [row][wave]
    __shared__ float lds_m[16];        // combined row maxes
    __shared__ int   lds_mask[16];     // mask row for current step

    const int tid  = threadIdx.x;
    const int wave = tid >> 5;             // 0..2 : output column tile
    const int lane = tid & 31;
    const int half = lane >> 4;            // 0 or 1
    const int l    = lane & 15;
    const int b0   = blockIdx.x * 16;      // first batch row of this tile

    // phase-A mapping: row r, 8-col segment k
    const int ar = tid / 6;                // 0..15
    const int ak = tid - 6 * ar;           // 0..5

    // Preload B operand: Q = exp(trans), this wave's 48x16 column block.
    // B layout (4x16 f32, mirrors documented A layout): VGPR0 = K=2*half row,
    // VGPR1 = K=2*half+1 row, N = lane%16.
    v2f bq[12];
    #pragma unroll
    for (int kk = 0; kk < 12; ++kk) {
        const int krow = kk * 4 + 2 * half;
        const int col  = wave * 16 + l;
        bq[kk].x = fast_exp(trans[krow * T_SZ + col]);
        bq[kk].y = fast_exp(trans[(krow + 1) * T_SZ + col]);
    }

    // init: score[r][c] = start[c] + em[0][b0+r][c]
    for (int e = tid; e < 16 * T_SZ; e += 96) {
        const int r = e / T_SZ, c = e % T_SZ;
        lds_S[r * SPAD + c] = start_tr[c] + emissions[(b0 + r) * T_SZ + c];
    }
    __syncthreads();
    // init partial row maxes [row][16-col segment]
    if (tid < 48) {
        const int r = tid / 3, w = tid - 3 * r;
        float m = lds_S[r * SPAD + w * 16];
        #pragma unroll
        for (int c = 1; c < 16; ++c)
            m = fmaxf(m, lds_S[r * SPAD + w * 16 + c]);
        lds_pm[r * 4 + w] = m;
    }
    __syncthreads();

    // register-carried "old" scores of this wave's own C-tile
    // (C/D layout: row = v + 8*half, col = wave*16 + lane%16)
    float prev[8];
    #pragma unroll
    for (int v = 0; v < 8; ++v)
        prev[v] = lds_S[(v + 8 * half) * SPAD + wave * 16 + l];

    // advancing pointers (stride folded per step; row offsets fold into IOFFSET)
    const float* emp   = emissions + (size_t)(B_SZ * T_SZ) + (b0 * T_SZ) + wave * 16 + l;
    const int*   maskp = mask + B_SZ + b0;

    for (int s = 1; s < S_LEN; ++s) {
        // ---- phase A: combine maxes, publish m & mask, build P ----
        {
            const float m = fmaxf(fmaxf(lds_pm[ar * 4 + 0], lds_pm[ar * 4 + 1]),
                                  lds_pm[ar * 4 + 2]);
            if (ak == 0) lds_m[ar] = m;
            if (ak == 5) lds_mask[ar] = maskp[ar];
            const float4 s0 = *(const float4*)&lds_S[ar * SPAD + ak * 8];
            const float4 s1 = *(const float4*)&lds_S[ar * SPAD + ak * 8 + 4];
            float4 p0, p1;
            p0.x = fast_exp(s0.x - m); p0.y = fast_exp(s0.y - m);
            p0.z = fast_exp(s0.z - m); p0.w = fast_exp(s0.w - m);
            p1.x = fast_exp(s1.x - m); p1.y = fast_exp(s1.y - m);
            p1.z = fast_exp(s1.z - m); p1.w = fast_exp(s1.w - m);
            *(float4*)&lds_P[ar * PPAD + ak * 8]     = p0;
            *(float4*)&lds_P[ar * PPAD + ak * 8 + 4] = p1;
        }
        __syncthreads();

        // ---- phase B: 16x48 @ 48x16 GEMM via V_WMMA_F32_16X16X4_F32 ----
        // Two independent accumulator chains to hide WMMA D->C latency.
        v8f acc0 = {}, acc1 = {};
        #pragma unroll
        for (int kk = 0; kk < 12; kk += 2) {
            v2f a0, a1;
            const int kb0 = kk * 4 + 2 * half;        // A: M = l, K = kb, kb+1
            const int kb1 = (kk + 1) * 4 + 2 * half;
            a0.x = lds_P[l * PPAD + kb0];
            a0.y = lds_P[l * PPAD + kb0 + 1];
            a1.x = lds_P[l * PPAD + kb1];
            a1.y = lds_P[l * PPAD + kb1 + 1];
            acc0 = __builtin_amdgcn_wmma_f32_16x16x4_f32(
                false, a0, false, bq[kk], (short)0, acc0, false, false);
            acc1 = __builtin_amdgcn_wmma_f32_16x16x4_f32(
                false, a1, false, bq[kk + 1], (short)0, acc1, false, false);
        }
        // epilogue: next = em + m + ln2*log2(PQ), masked keep-old (registers).
        float nv[8];
        #pragma unroll
        for (int v = 0; v < 8; ++v) {
            const int row = v + 8 * half;
            const float em  = emp[row * T_SZ];
            const float nxt = fmaf(LN2, __builtin_amdgcn_logf(acc0[v] + acc1[v]),
                                   em + lds_m[row]);
            const float sel = (lds_mask[row] != 0) ? nxt : prev[v];
            lds_S[row * SPAD + wave * 16 + l] = sel;
            prev[v] = sel;
            nv[v]   = sel;
        }
        // in-register partial row max over this wave's 16 cols (half-wave xor)
        #pragma unroll
        for (int v = 0; v < 8; ++v) {
            float mv = nv[v];
            mv = fmaxf(mv, __shfl_xor(mv, 1, 32));
            mv = fmaxf(mv, __shfl_xor(mv, 2, 32));
            mv = fmaxf(mv, __shfl_xor(mv, 4, 32));
            mv = fmaxf(mv, __shfl_xor(mv, 8, 32));
            if (l == 0) lds_pm[(v + 8 * half) * 4 + wave] = mv;
        }
        __syncthreads();

        emp   += B_SZ * T_SZ;
        maskp += B_SZ;
    }

    // final: den[b] = logsumexp_c( score[r][c] + end[c] )
    if (tid < 16) {
        const int r = tid;
        float m = -1e30f;
        for (int c = 0; c < T_SZ; ++c)
            m = fmaxf(m, lds_S[r * SPAD + c] + end_tr[c]);
        float sum = 0.f;
        for (int c = 0; c < T_SZ; ++c)
            sum += fast_exp(lds_S[r * SPAD + c] + end_tr[c] - m);
        den[b0 + r] = m + fast_log(sum);
    }
}

// ---------------------------------------------------------------------------
// Numerator: gold-path score. One thread per batch element.
// ---------------------------------------------------------------------------
__global__ __launch_bounds__(256) void crf_score_kernel(
    const float*     __restrict__ emissions,
    const long long* __restrict__ tags,     // (S,B) int64
    const int*       __restrict__ mask,     // (S,B) int32
    const float*     __restrict__ start_tr,
    const float*     __restrict__ end_tr,
    const float*     __restrict__ trans,
    float* __restrict__ num)                // (B)
{
    const int b = threadIdx.x;
    int t0 = (int)tags[b];
    float sc = start_tr[t0] + emissions[b * T_SZ + t0];
    int prev = t0;       // raw previous tag (reference uses tags[:-1] unmasked)
    int last = t0;       // last tag at a masked step
    for (int s = 1; s < S_LEN; ++s) {
        const int tg = (int)tags[(size_t)s * B_SZ + b];
        const int m  = mask[s * B_SZ + b];
        if (m) {
            sc += trans[prev * T_SZ + tg]
                + emissions[(size_t)s * (B_SZ * T_SZ) + b * T_SZ + tg];
            last = tg;
        }
        prev = tg;
    }
    sc += end_tr[last];
    num[b] = sc;
}

// ---------------------------------------------------------------------------
// Final reduce: sum_b (num[b] - den[b]) -> scalar
// ---------------------------------------------------------------------------
__global__ __launch_bounds__(256) void crf_reduce_kernel(
    const float* __restrict__ num, const float* __restrict__ den,
    float* __restrict__ out)
{
    __shared__ float buf[256];
    const int t = threadIdx.x;
    buf[t] = num[t] - den[t];
    __syncthreads();
    for (int off = 128; off > 0; off >>= 1) {
        if (t < off) buf[t] += buf[t + off];
        __syncthreads();
    }
    if (t == 0) out[0] = buf[0];
}

extern "C" void kernel_launch(void* const* d_in, const int* in_sizes, int n_in,
                              void* d_out, int out_size, void* d_ws, size_t ws_size,
                              hipStream_t stream) {
    (void)in_sizes; (void)n_in; (void)out_size; (void)ws_size;
    const float*     em    = (const float*)d_in[0];
    const long long* tags  = (const long long*)d_in[1];
    const int*       mask  = (const int*)d_in[2];
    const float*     st    = (const float*)d_in[3];
    const float*     en    = (const float*)d_in[4];
    const float*     tr    = (const float*)d_in[5];

    float* num = (float*)d_ws;
    float* den = num + B_SZ;

    crf_score_kernel<<<1, 256, 0, stream>>>(em, tags, mask, st, en, tr, num);
    crf_forward_norm_kernel<<<B_SZ / 16, 96, 0, stream>>>(em, mask, st, en, tr, den);
    crf_reduce_kernel<<<1, 256, 0, stream>>>(num, den, (float*)d_out);
}